// HMMModel_4234837753911
// MI455X (gfx1250) — compile-verified
//
#include <hip/hip_runtime.h>
#include <math.h>

#define TT 8192
#define NN 2048
static constexpr float kLog2Pi = 1.8378770664093453f;

typedef float v2f __attribute__((ext_vector_type(2)));
typedef float v8f __attribute__((ext_vector_type(8)));

// ---------------------------------------------------------------------------
// Kernel 1: per-state Gaussian parameters -> GEMM operands
//   B[0][n]=mu0/v0  B[1][n]=mu1/v1  B[2][n]=-1/(2v0)  B[3][n]=-1/(2v1)
//   bias[n] = -0.5*(2*log2pi + log v0 + log v1) - 0.5*(mu0^2/v0 + mu1^2/v1)
// ---------------------------------------------------------------------------
__global__ void hmm_prep_kernel(const float* __restrict__ mean,
                                const float* __restrict__ stdv,
                                float* __restrict__ Bmat,
                                float* __restrict__ bias) {
    int j = blockIdx.x * blockDim.x + threadIdx.x;
    if (j >= NN) return;
    float m0 = mean[2 * j], m1 = mean[2 * j + 1];
    float s0 = stdv[2 * j], s1 = stdv[2 * j + 1];
    float v0 = s0 * s0, v1 = s1 * s1;
    Bmat[0 * NN + j] = m0 / v0;
    Bmat[1 * NN + j] = m1 / v1;
    Bmat[2 * NN + j] = -0.5f / v0;
    Bmat[3 * NN + j] = -0.5f / v1;
    bias[j] = -0.5f * (2.0f * kLog2Pi + logf(v0) + logf(v1))
              - 0.5f * (m0 * m0 / v0 + m1 * m1 / v1);
}

// ---------------------------------------------------------------------------
// Kernel 2: logT = log(trans), elementwise over N*N
// ---------------------------------------------------------------------------
__global__ void hmm_logt_kernel(const float* __restrict__ trans,
                                float* __restrict__ logT) {
    int i = blockIdx.x * blockDim.x + threadIdx.x;
    logT[i] = logf(trans[i]);
}

// ---------------------------------------------------------------------------
// Kernel 3: emission log-likelihood via V_WMMA_F32_16X16X4_F32.
//   e_ll[t,n] = A[t,:] (1x4) dot B[:,n] (4x1) + bias[n],
//   A[t,:] = [x0, x1, x0^2, x1^2].
// One wave per 16(t) x 16(n) tile. A layout (ISA 7.12.2, 32-bit A 16x4):
//   lanes 0-15 supply K=0 (VGPR0) / K=1 (VGPR1); lanes 16-31 K=2 / K=3.
// B (4x16): lanes 0-15 rows K=0/K=1; lanes 16-31 rows K=2/K=3.
// C/D: VGPR r, lane l -> row M = r + 8*(l>=16), col = l&15.
// ---------------------------------------------------------------------------
__global__ void __launch_bounds__(256)
hmm_emis_kernel(const float* __restrict__ ev,
                const float* __restrict__ Bmat,
                const float* __restrict__ bias,
                float* __restrict__ score) {
    const int tilesN = NN / 16;                       // 128
    int wave = blockIdx.x * (blockDim.x >> 5) + (threadIdx.x >> 5);
    int lane = threadIdx.x & 31;
    int tile_n = wave % tilesN;
    int tile_t = wave / tilesN;
    int t0 = tile_t * 16, n0 = tile_n * 16;
    int l = lane & 15;
    int half = lane >> 4;                             // 0: K=0,1   1: K=2,3

    float x0 = ev[(size_t)(t0 + l) * 2 + 0];
    float x1 = ev[(size_t)(t0 + l) * 2 + 1];
    v2f A;
    A[0] = half ? x0 * x0 : x0;                       // K = 2*half
    A[1] = half ? x1 * x1 : x1;                       // K = 2*half + 1
    v2f B;
    B[0] = Bmat[(size_t)(half * 2 + 0) * NN + n0 + l];
    B[1] = Bmat[(size_t)(half * 2 + 1) * NN + n0 + l];
    float c = bias[n0 + l];
    v8f C;
#pragma unroll
    for (int r = 0; r < 8; ++r) C[r] = c;

    v8f D = __builtin_amdgcn_wmma_f32_16x16x4_f32(
        /*neg_a=*/false, A, /*neg_b=*/false, B,
        /*c_mod=*/(short)0, C, /*reuse_a=*/false, /*reuse_b=*/false);

#pragma unroll
    for (int r = 0; r < 8; ++r)
        score[(size_t)(t0 + r + half * 8) * NN + n0 + l] = D[r];
}

// ---------------------------------------------------------------------------
// Kernel 4: nl[0] = log(prior) + e_ll[0]   (in place on score row 0)
// ---------------------------------------------------------------------------
__global__ void hmm_init_kernel(const float* __restrict__ prior,
                                float* __restrict__ score) {
    int j = blockIdx.x * blockDim.x + threadIdx.x;
    if (j < NN) score[j] = logf(prior[j]) + score[j];
}

// ---------------------------------------------------------------------------
// Device-scope grid barrier (16 co-resident blocks) for the sequential scan.
// ---------------------------------------------------------------------------
__device__ __forceinline__ void grid_sync(int* cnt, int* gen, int nblocks) {
    __syncthreads();
    if (threadIdx.x == 0) {
        int g = __hip_atomic_load(gen, __ATOMIC_RELAXED, __HIP_MEMORY_SCOPE_AGENT);
        int p = __hip_atomic_fetch_add(cnt, 1, __ATOMIC_ACQ_REL, __HIP_MEMORY_SCOPE_AGENT);
        if (p == nblocks - 1) {
            __hip_atomic_store(cnt, 0, __ATOMIC_RELAXED, __HIP_MEMORY_SCOPE_AGENT);
            __hip_atomic_fetch_add(gen, 1, __ATOMIC_RELEASE, __HIP_MEMORY_SCOPE_AGENT);
        } else {
            while (__hip_atomic_load(gen, __ATOMIC_ACQUIRE, __HIP_MEMORY_SCOPE_AGENT) == g)
                __builtin_amdgcn_s_sleep(1);
        }
    }
    __syncthreads();
}

// ---------------------------------------------------------------------------
// Kernel 5: Viterbi forward scan (persistent cooperative kernel).
// 16 blocks x 1024 threads. Block wg owns columns [wg*128, wg*128+128).
// Thread (jg = tid&31, slice = tid>>5): 4 columns j0..j0+3, k-slice of 64.
// logT kept [k][j]: per k, 32 lanes read 4 consecutive floats each ->
// fully coalesced 512B per wave; prev[k] broadcast from LDS.
// ---------------------------------------------------------------------------
__global__ void __launch_bounds__(1024)
hmm_fwd_kernel(const float* __restrict__ logT,
               float* __restrict__ score,
               int* __restrict__ sync) {
    __shared__ float prev[NN];              // 8 KB
    __shared__ float4 red[32][32];          // 16 KB: [slice][jgroup]
    const int tid = threadIdx.x;
    const int jg = tid & 31;
    const int slice = tid >> 5;
    const int j0 = blockIdx.x * 128 + jg * 4;
    const float* colbase = logT + j0;
    const int kbeg = slice * 64;

    for (int t = 1; t < TT; ++t) {
        // stage nl[t-1] into LDS (broadcast source for every lane)
        const float* prow = score + (size_t)(t - 1) * NN;
        prev[tid] = prow[tid];
        prev[tid + 1024] = prow[tid + 1024];
        __syncthreads();

        float4 m;
        m.x = m.y = m.z = m.w = -3.0e38f;
#pragma unroll 8
        for (int kk = 0; kk < 64; ++kk) {
            int k = kbeg + kk;
            if ((kk & 7) == 0)
                __builtin_prefetch(colbase + (size_t)(k + 16) * NN, 0, 0);
            const float4 w = *reinterpret_cast<const float4*>(colbase + (size_t)k * NN);
            float p = prev[k];
            m.x = fmaxf(m.x, p + w.x);
            m.y = fmaxf(m.y, p + w.y);
            m.z = fmaxf(m.z, p + w.z);
            m.w = fmaxf(m.w, p + w.w);
        }
        red[slice][jg] = m;
        __syncthreads();

        if (slice == 0) {
            float4 mm = red[0][jg];
#pragma unroll
            for (int s = 1; s < 32; ++s) {
                float4 o = red[s][jg];
                mm.x = fmaxf(mm.x, o.x);
                mm.y = fmaxf(mm.y, o.y);
                mm.z = fmaxf(mm.z, o.z);
                mm.w = fmaxf(mm.w, o.w);
            }
            // in-place: e_ll row t -> nl row t
            float* row = score + (size_t)t * NN + j0;
            float4 e = *reinterpret_cast<const float4*>(row);
            float4 r;
            r.x = mm.x + e.x; r.y = mm.y + e.y;
            r.z = mm.z + e.z; r.w = mm.w + e.w;
            *reinterpret_cast<float4*>(row) = r;
        }
        __threadfence();
        grid_sync(sync, sync + 1, (int)gridDim.x);
    }
}

// ---------------------------------------------------------------------------
// Kernel 6: backtrace, single persistent block.
//   out[T-1]=argmax nl[T-1]; out[T]=max nl[T-1];
//   out[t] = argmax_j nl[t,j] + logT[j, next]   (first-index tie-break)
// ---------------------------------------------------------------------------
__global__ void __launch_bounds__(1024)
hmm_bwd_kernel(const float* __restrict__ score,
               const float* __restrict__ logT,
               float* __restrict__ out) {
    __shared__ float sv[1024];
    __shared__ int si[1024];
    __shared__ int nextState;
    const int tid = threadIdx.x;

    // terminal argmax + probability
    {
        const float* row = score + (size_t)(TT - 1) * NN;
        float v = row[tid];
        int idx = tid;
        float v2 = row[tid + 1024];
        if (v2 > v) { v = v2; idx = tid + 1024; }
        sv[tid] = v; si[tid] = idx;
        __syncthreads();
        for (int s = 512; s > 0; s >>= 1) {
            if (tid < s) {
                float ov = sv[tid + s]; int oi = si[tid + s];
                if (ov > sv[tid] || (ov == sv[tid] && oi < si[tid])) {
                    sv[tid] = ov; si[tid] = oi;
                }
            }
            __syncthreads();
        }
        if (tid == 0) {
            out[TT - 1] = (float)si[0];
            out[TT] = sv[0];
            nextState = si[0];
        }
        __syncthreads();
    }

    for (int t = TT - 2; t >= 0; --t) {
        int ns = nextState;
        const float* row = score + (size_t)t * NN;
        float a = row[tid] + logT[(size_t)tid * NN + ns];
        float b = row[tid + 1024] + logT[(size_t)(tid + 1024) * NN + ns];
        float v = a; int idx = tid;
        if (b > a) { v = b; idx = tid + 1024; }
        sv[tid] = v; si[tid] = idx;
        __syncthreads();
        for (int s = 512; s > 0; s >>= 1) {
            if (tid < s) {
                float ov = sv[tid + s]; int oi = si[tid + s];
                if (ov > sv[tid] || (ov == sv[tid] && oi < si[tid])) {
                    sv[tid] = ov; si[tid] = oi;
                }
            }
            __syncthreads();
        }
        if (tid == 0) {
            out[t] = (float)si[0];
            nextState = si[0];
        }
        __syncthreads();
    }
}

// ---------------------------------------------------------------------------
extern "C" void kernel_launch(void* const* d_in, const int* in_sizes, int n_in,
                              void* d_out, int out_size, void* d_ws, size_t ws_size,
                              hipStream_t stream) {
    const float* evidence  = (const float*)d_in[0];   // [T,2]
    const float* prior     = (const float*)d_in[1];   // [N]
    const float* trans     = (const float*)d_in[2];   // [N,N]
    const float* emis_mean = (const float*)d_in[3];   // [N,2]
    const float* emis_std  = (const float*)d_in[4];   // [N,2]
    float* out = (float*)d_out;                       // [T path + 1 prob]

    float* score = (float*)d_ws;                      // T*N   (e_ll -> nl, in place)
    float* logT  = score + (size_t)TT * NN;           // N*N
    float* Bmat  = logT + (size_t)NN * NN;            // 4*N
    float* bias  = Bmat + 4 * NN;                     // N
    int*   sync  = (int*)(bias + NN);                 // 2 ints

    hipMemsetAsync(sync, 0, 2 * sizeof(int), stream);

    hmm_prep_kernel<<<NN / 256, 256, 0, stream>>>(emis_mean, emis_std, Bmat, bias);
    hmm_logt_kernel<<<(NN * NN) / 256, 256, 0, stream>>>(trans, logT);

    const int waves = (TT / 16) * (NN / 16);          // 65536 tiles
    hmm_emis_kernel<<<waves / 8, 256, 0, stream>>>(evidence, Bmat, bias, score);

    hmm_init_kernel<<<NN / 256, 256, 0, stream>>>(prior, score);

    hmm_fwd_kernel<<<16, 1024, 0, stream>>>(logT, score, sync);

    hmm_bwd_kernel<<<1, 1024, 0, stream>>>(score, logT, out);
}